// HR2dHk_37417755082908
// MI455X (gfx1250) — compile-verified
//
#include <hip/hip_runtime.h>
#include <cstdint>
#include <cstddef>

// ---------------------------------------------------------------------------
// HR -> dH(k) builder for CDNA5 (gfx1250).
//   out[d][k] = dH + dH^H,  dH = blockdiag(onsite) + scatter(edge vals)
// Memory-bound: 384 MB output. Strategy:
//   K1 zero-fill (non-temporal 128-bit stores; don't thrash 192MB L2)
//   K2 phases: k.R via V_WMMA_F32_16X16X4_F32 (one wave / 16 edges);
//      exact lattice periodicity reduction + sinpif/cospif -> sin,cos in ws
//   K3 diagonal blocks: onsite + onsite^T (plain stores, race-free)
//   K4 edge scatter: global_atomic_add_f32 at (i,j) and conj at (j,i)
// ---------------------------------------------------------------------------

typedef float v2f __attribute__((ext_vector_type(2)));
typedef float v4f __attribute__((ext_vector_type(4)));
typedef float v8f __attribute__((ext_vector_type(8)));

#define FEATN 68
#define NORB  10

// inverse map: feature f -> (row a, col b, factor)
__constant__ int c_AOF[FEATN] = {
  0,0,0,0,0,0,0,0,0,0,
  1,1,1,1,1,1,1,1,1,
  2,2,2,3,3,3,4,4,4,
  2,2,2,2,2,3,3,3,3,3,4,4,4,4,4,
  5,5,5,5,5,6,6,6,6,6,7,7,7,7,7,8,8,8,8,8,9,9,9,9,9};
__constant__ int c_BOF[FEATN] = {
  0,1,2,3,4,5,6,7,8,9,
  1,2,3,4,5,6,7,8,9,
  2,3,4,2,3,4,2,3,4,
  5,6,7,8,9,5,6,7,8,9,5,6,7,8,9,
  5,6,7,8,9,5,6,7,8,9,5,6,7,8,9,5,6,7,8,9,5,6,7,8,9};
__constant__ float c_FAC[FEATN] = {
  0.5f,1,1,1,1,1,1,1,1,1,
  0.5f,1,1,1,1,1,1,1,1,
  0.5f,0.5f,0.5f,0.5f,0.5f,0.5f,0.5f,0.5f,0.5f,
  1,1,1,1,1,1,1,1,1,1,1,1,1,1,1,
  0.5f,0.5f,0.5f,0.5f,0.5f,0.5f,0.5f,0.5f,0.5f,0.5f,
  0.5f,0.5f,0.5f,0.5f,0.5f,0.5f,0.5f,0.5f,0.5f,0.5f,
  0.5f,0.5f,0.5f,0.5f,0.5f};

// forward map: (a,b) -> feature index (-1 = structurally zero), and factor
__constant__ int c_FIDX[NORB * NORB] = {
   0,  1,  2,  3,  4,  5,  6,  7,  8,  9,
  -1, 10, 11, 12, 13, 14, 15, 16, 17, 18,
  -1, -1, 19, 20, 21, 28, 29, 30, 31, 32,
  -1, -1, 22, 23, 24, 33, 34, 35, 36, 37,
  -1, -1, 25, 26, 27, 38, 39, 40, 41, 42,
  -1, -1, -1, -1, -1, 43, 44, 45, 46, 47,
  -1, -1, -1, -1, -1, 48, 49, 50, 51, 52,
  -1, -1, -1, -1, -1, 53, 54, 55, 56, 57,
  -1, -1, -1, -1, -1, 58, 59, 60, 61, 62,
  -1, -1, -1, -1, -1, 63, 64, 65, 66, 67};
__constant__ float c_FFAC[NORB * NORB] = {
  0.5f,1,1,1,1,1,1,1,1,1,
  1,0.5f,1,1,1,1,1,1,1,1,
  1,1,0.5f,0.5f,0.5f,1,1,1,1,1,
  1,1,0.5f,0.5f,0.5f,1,1,1,1,1,
  1,1,0.5f,0.5f,0.5f,1,1,1,1,1,
  1,1,1,1,1,0.5f,0.5f,0.5f,0.5f,0.5f,
  1,1,1,1,1,0.5f,0.5f,0.5f,0.5f,0.5f,
  1,1,1,1,1,0.5f,0.5f,0.5f,0.5f,0.5f,
  1,1,1,1,1,0.5f,0.5f,0.5f,0.5f,0.5f,
  1,1,1,1,1,0.5f,0.5f,0.5f,0.5f,0.5f};

// ---------------------------------------------------------------------------
// K1: zero-fill output (n4 16-byte elements), grid-stride, non-temporal
__global__ void hr_zero_f4(v4f* __restrict__ p, size_t n4) {
  size_t i = (size_t)blockIdx.x * blockDim.x + threadIdx.x;
  size_t stride = (size_t)gridDim.x * blockDim.x;
  const v4f z = {0.f, 0.f, 0.f, 0.f};
  for (; i < n4; i += stride)
    __builtin_nontemporal_store(z, &p[i]);
}

// ---------------------------------------------------------------------------
// K2: phases. One wave per tile of 16 edges.
//   D[16x16] = A[16x4] x B[4x16]; A rows = k-points (rows >= NK zeroed, K=3
//   padded with 0), B cols = edge_cell_shift^T. D[k][n] = kpoints[k].R[e0+n].
// 32-bit A 16x4 layout: lane<16 holds M=lane, {v0=K0, v1=K1}; lane>=16 holds
// M=lane-16, {v0=K2, v1=K3}. B 4x16: v0 = rows K0|K2, v1 = rows K1|K3.
// All loads are unconditional (clamped indices) so no exec divergence occurs
// before the WMMA; selection is done with v_cndmask.
__global__ void hr_phase_wmma(const float* __restrict__ kp,
                              const float* __restrict__ shift,
                              float* __restrict__ Sph, float* __restrict__ Cph,
                              int E, int NK) {
  int lane = threadIdx.x & 31;
  int tile = blockIdx.x;                 // blockDim == 32 -> one wave per block
  int e0 = tile * 16;
  int m = lane & 15;
  bool lo = lane < 16;
  bool vm = m < NK;

  int mk = vm ? m : 0;                   // clamped row for unconditional load
  float k0 = kp[mk * 3 + 0];
  float k1 = kp[mk * 3 + 1];
  float k2 = kp[mk * 3 + 2];
  float a0 = vm ? (lo ? k0 : k2) : 0.f;
  float a1 = (vm && lo) ? k1 : 0.f;

  int e = e0 + m;
  if (e >= E) e = E - 1;                 // clamp (invalid columns never stored)
  float s0 = shift[e * 3 + 0];
  float s1 = shift[e * 3 + 1];
  float s2 = shift[e * 3 + 2];
  float b0 = lo ? s0 : s2;
  float b1 = lo ? s1 : 0.f;

  v2f A;  A.x = a0; A.y = a1;
  v2f Bv; Bv.x = b0; Bv.y = b1;
  v8f Cz = {0.f, 0.f, 0.f, 0.f, 0.f, 0.f, 0.f, 0.f};
  // D = A x B + 0   ->  v_wmma_f32_16x16x4_f32
  v8f D = __builtin_amdgcn_wmma_f32_16x16x4_f32(
      false, A, false, Bv, (short)0, Cz, false, false);

  // C/D layout: lane<16 -> row M = vgpr index. We need rows 0..NK-1 (<=4).
  if (lo && (e0 + m) < E) {
#pragma unroll
    for (int r = 0; r < 4; ++r) {
      if (r < NK) {
        float dot = D[r];                // k.R, R integer lattice shift
        // exp(-i*2pi*dot): exact periodicity -> reduce by nearest integer.
        float fr = dot - rintf(dot);     // |fr| <= 0.5, reduction is exact
        // i*phase = sin(2pi fr) + i*cos(2pi fr); sinpi/cospi: no slow path
        Sph[(size_t)r * E + e0 + m] = sinpif(2.f * fr);
        Cph[(size_t)r * E + e0 + m] = cospif(2.f * fr);
      }
    }
  }
}

// ---------------------------------------------------------------------------
// K3: hermitianized diagonal blocks: value(a,b) = blk(a,b) + blk(b,a), real.
// One thread per (n,a,b); writes 3*NK planes. Race-free -> plain stores.
__global__ void hr_diag(const float* __restrict__ nf, float* __restrict__ out,
                        int N, int NK) {
  int t = blockIdx.x * blockDim.x + threadIdx.x;
  int total = N * NORB * NORB;
  if (t >= total) return;
  int n = t / (NORB * NORB);
  int ab = t - n * (NORB * NORB);
  int a = ab / NORB, b = ab - a * NORB;

  float v = 0.f;
  int fab = c_FIDX[a * NORB + b];
  if (fab >= 0) v += c_FFAC[a * NORB + b] * nf[(size_t)n * FEATN + fab];
  int fba = c_FIDX[b * NORB + a];
  if (fba >= 0) v += c_FFAC[b * NORB + a] * nf[(size_t)n * FEATN + fba];

  size_t NG = (size_t)N * NORB;
  size_t mat = NG * NG;
  size_t off = (size_t)(n * NORB + a) * NG + (size_t)(n * NORB + b);
  int planes = 3 * NK;
  for (int dk = 0; dk < planes; ++dk)
    out[2 * ((size_t)dk * mat + off)] = v;  // real part; imag stays 0
}

// ---------------------------------------------------------------------------
// K4: edge scatter. One thread per (edge, feature element).
//   val = (sin + i*cos) * (-ev[d]) * fac * feat  at (i,j); conj at (j,i).
__global__ void hr_edge_scatter(const float* __restrict__ ef,
                                const float* __restrict__ ev,
                                const int* __restrict__ eidx,
                                const float* __restrict__ Sph,
                                const float* __restrict__ Cph,
                                float* __restrict__ out,
                                int E, int N, int NK) {
  int t = blockIdx.x * blockDim.x + threadIdx.x;
  if (t >= E * FEATN) return;
  int e = t / FEATN;
  int f = t - e * FEATN;
  int a = c_AOF[f], b = c_BOF[f];
  float h = c_FAC[f] * ef[t];            // ef[e*FEATN + f]

  int i = eidx[e];
  int j = eidx[E + e];

  size_t NG = (size_t)N * NORB;
  size_t mat = NG * NG;
  size_t rij = (size_t)(i * NORB + a) * NG + (size_t)(j * NORB + b);
  size_t rji = (size_t)(j * NORB + b) * NG + (size_t)(i * NORB + a);

  float sv[8], cv[8];
  int nk = NK > 8 ? 8 : NK;
  for (int k = 0; k < nk; ++k) {
    sv[k] = Sph[(size_t)k * E + e];
    cv[k] = Cph[(size_t)k * E + e];
  }

#pragma unroll
  for (int d = 0; d < 3; ++d) {
    float scale = -ev[(size_t)e * 3 + d] * h;
    for (int k = 0; k < nk; ++k) {
      size_t base = (size_t)(d * NK + k) * mat;
      float re = sv[k] * scale;
      float im = cv[k] * scale;
      atomicAdd(out + 2 * (base + rij) + 0, re);
      atomicAdd(out + 2 * (base + rij) + 1, im);
      atomicAdd(out + 2 * (base + rji) + 0, re);   // conj-transpose mirror
      atomicAdd(out + 2 * (base + rji) + 1, -im);
    }
  }
}

// ---------------------------------------------------------------------------
extern "C" void kernel_launch(void* const* d_in, const int* in_sizes, int n_in,
                              void* d_out, int out_size, void* d_ws, size_t ws_size,
                              hipStream_t stream) {
  const float* ef    = (const float*)d_in[0];  // edge_features   [E,68]
  const float* nf    = (const float*)d_in[1];  // node_features   [N,68]
  const float* ev    = (const float*)d_in[2];  // edge_vectors    [E,3]
  const float* shift = (const float*)d_in[3];  // edge_cell_shift [E,3]
  const float* kp    = (const float*)d_in[4];  // kpoints         [NK,3]
  const int*   eidx  = (const int*)d_in[5];    // edge_index      [2,E]

  const int E  = in_sizes[0] / FEATN;
  const int N  = in_sizes[1] / FEATN;
  const int NK = in_sizes[4] / 3;

  float* out = (float*)d_out;
  size_t NG = (size_t)N * NORB;
  size_t totalFloats = (size_t)3 * NK * NG * NG * 2;  // complex64 interleaved

  float* Sph = (float*)d_ws;                 // [NK, E] sin
  float* Cph = Sph + (size_t)NK * E;         // [NK, E] cos

  // K1: zero output (NG even -> totalFloats divisible by 4)
  hr_zero_f4<<<4096, 256, 0, stream>>>((v4f*)out, totalFloats / 4);

  // K2: phases via WMMA (one wave per 16-edge tile)
  hr_phase_wmma<<<(E + 15) / 16, 32, 0, stream>>>(kp, shift, Sph, Cph, E, NK);

  // K3: diagonal blocks
  hr_diag<<<(N * NORB * NORB + 255) / 256, 256, 0, stream>>>(nf, out, N, NK);

  // K4: edge scatter (after zero+diag by stream order)
  hr_edge_scatter<<<(E * FEATN + 255) / 256, 256, 0, stream>>>(
      ef, ev, eidx, Sph, Cph, out, E, N, NK);
}